// GCNConv_88424786690100
// MI455X (gfx1250) — compile-verified
//
#include <hip/hip_runtime.h>
#include <hip/hip_bf16.h>

typedef __attribute__((ext_vector_type(2))) float v2f;
typedef __attribute__((ext_vector_type(8))) float v8f;

#define D 128  // D_IN == D_OUT == 128

// ---------------- degree kernels ----------------
__global__ void deg_init_kernel(float* deg, int n) {
    int i = blockIdx.x * blockDim.x + threadIdx.x;
    if (i < n) deg[i] = 1.0f;  // self-loop contributes 1 to every node
}

__global__ void deg_count_kernel(const int* row, float* deg, int E) {
    int e = blockIdx.x * blockDim.x + threadIdx.x;
    if (e < E) atomicAdd(&deg[row[e]], 1.0f);
}

__global__ void deg_rsqrt_kernel(const float* deg, float* dis, int n) {
    int i = blockIdx.x * blockDim.x + threadIdx.x;
    if (i < n) dis[i] = rsqrtf(deg[i]);  // deg >= 1, never inf
}

// ---------------- fp32 WMMA GEMM: h = x @ W ----------------
// One wave computes one 16x16 tile of h. blockDim = 256 (8 waves); wave w
// handles N-tile w (D_OUT/16 == 8), block b handles M-tile b.
__global__ void gemm_wmma_f32(const float* __restrict__ x,
                              const float* __restrict__ W,
                              float* __restrict__ h) {
    const int wave = threadIdx.x >> 5;        // 0..7 -> N tile
    const int lane = threadIdx.x & 31;
    const int half = lane >> 4;               // 0: lanes 0-15, 1: lanes 16-31
    const int idx  = lane & 15;
    const int m0 = blockIdx.x << 4;
    const int n0 = wave << 4;

    v8f c = {};
    // A 16x4 (MxK): v0 = K = 2*half, v1 = K = 2*half+1, M = idx
    // B 4x16 (KxN): v0 = K = 2*half, v1 = K = 2*half+1, N = idx
    const float* xr = x + (m0 + idx) * D;
    const float* Wc = W + idx + n0;
#pragma unroll 8
    for (int k0 = 0; k0 < D; k0 += 4) {
        v2f a, b;
        a.x = xr[k0 + 2 * half + 0];
        a.y = xr[k0 + 2 * half + 1];
        b.x = Wc[(k0 + 2 * half + 0) * D];
        b.y = Wc[(k0 + 2 * half + 1) * D];
        c = __builtin_amdgcn_wmma_f32_16x16x4_f32(
            /*neg_a=*/false, a, /*neg_b=*/false, b,
            /*c_mod=*/(short)0, c, /*reuse_a=*/false, /*reuse_b=*/false);
    }
    // C/D layout: VGPR g, lane -> row m0 + g + 8*half, col n0 + idx
    float* hp = h + (m0 + 8 * half) * D + n0 + idx;
#pragma unroll
    for (int g = 0; g < 8; ++g) hp[g * D] = c[g];
}

// ---------------- out = bias + h * dis^2 (self-loop term) ----------------
__global__ void self_bias_kernel(const float* __restrict__ h,
                                 const float* __restrict__ dis,
                                 const float* __restrict__ bias,
                                 float* __restrict__ out, int n) {
    int gid = blockIdx.x * blockDim.x + threadIdx.x;   // one thread per (node, ch)
    if (gid >= n * D) return;
    int i = gid >> 7;          // node
    int d = gid & (D - 1);     // channel
    float s = dis[i];
    out[gid] = bias[d] + h[gid] * s * s;
}

// ---------------- edge scatter: out[col] += h[row] * w ----------------
// One wave per edge; each lane handles 4 channels (float4) -> coalesced
// 512B gather per edge + 4 global_atomic_add_f32 per lane. h/out live in L2.
__global__ void scatter_kernel(const int* __restrict__ row,
                               const int* __restrict__ col,
                               const float* __restrict__ h,
                               const float* __restrict__ dis,
                               float* __restrict__ out, int E) {
    int gid = blockIdx.x * blockDim.x + threadIdx.x;
    int e = gid >> 5;
    if (e >= E) return;
    int lane = gid & 31;
    int r = row[e];
    int c = col[e];
    float w = dis[r] * dis[c];
    const float4 hv = *(const float4*)(h + r * D + lane * 4);
    float* o = out + c * D + lane * 4;
    atomicAdd(o + 0, hv.x * w);
    atomicAdd(o + 1, hv.y * w);
    atomicAdd(o + 2, hv.z * w);
    atomicAdd(o + 3, hv.w * w);
}

extern "C" void kernel_launch(void* const* d_in, const int* in_sizes, int n_in,
                              void* d_out, int out_size, void* d_ws, size_t ws_size,
                              hipStream_t stream) {
    const float* x    = (const float*)d_in[0];
    const int*   ei   = (const int*)d_in[1];
    const float* W    = (const float*)d_in[2];
    const float* bias = (const float*)d_in[3];
    float* out = (float*)d_out;

    const int N = in_sizes[0] / D;       // 10000
    const int E = in_sizes[1] / 2;       // 640000
    const int* row = ei;
    const int* col = ei + E;

    // workspace layout: h [N*D] | deg [N] | dis [N]
    float* h   = (float*)d_ws;
    float* deg = h + (size_t)N * D;
    float* dis = deg + N;

    // 1) deg = 1 (self-loops)
    deg_init_kernel<<<(N + 255) / 256, 256, 0, stream>>>(deg, N);
    // 2) deg += count over row
    deg_count_kernel<<<(E + 255) / 256, 256, 0, stream>>>(row, deg, E);
    // 3) dis = deg^-1/2
    deg_rsqrt_kernel<<<(N + 255) / 256, 256, 0, stream>>>(deg, dis, N);
    // 4) h = x @ W   (fp32 WMMA; M tiles = N/16, N tiles = 8 waves/block)
    gemm_wmma_f32<<<N / 16, 256, 0, stream>>>(x, W, h);
    // 5) out = bias + h * dis^2
    self_bias_kernel<<<((size_t)N * D + 255) / 256, 256, 0, stream>>>(h, dis, bias, out, N);
    // 6) scatter edges: one wave per edge
    {
        long long threads = (long long)E * 32;
        int blocks = (int)((threads + 255) / 256);
        scatter_kernel<<<blocks, 256, 0, stream>>>(row, col, h, dis, out, E);
    }
}